// MaskGenerator_69252052680743
// MI455X (gfx1250) — compile-verified
//
#include <hip/hip_runtime.h>

typedef float v2f __attribute__((ext_vector_type(2)));
typedef float v8f __attribute__((ext_vector_type(8)));

#define TK 5

// ---------------------------------------------------------------------------
// Kernel 1: aligned_ctx = ctx_emb @ W_map^T via V_WMMA_F32_16X16X4_F32
// One wave32 per 16x16 output tile. aligned[m,n] = sum_k ctx[m,k]*Wmap[n,k]
//
// Fragment layouts (ISA 7.12.2, 32-bit):
//   A 16x4 : lane(0-15)=M, V0=K0 / V1=K1 ; lanes 16-31 carry K2/K3
//   B 4x16 : lane(0-15)=N, V0=K0 / V1=K1 ; lanes 16-31 carry K2/K3
//   C/D    : V[r] -> M = r (+8 for lanes 16-31), N = lane&15
// ---------------------------------------------------------------------------
__global__ void __launch_bounds__(32)
gemm_aligned_wmma(const float* __restrict__ ctx,   // [B,E] row-major
                  const float* __restrict__ Wmap,  // [E,E] row-major
                  float* __restrict__ aligned,     // [B,E]
                  int E)
{
    const int lane = threadIdx.x & 31;
    const int half = lane >> 4;       // 0: lanes 0-15, 1: lanes 16-31
    const int l    = lane & 15;

    const int n0 = blockIdx.x * 16;   // N tile (cols of aligned, rows of Wmap)
    const int m0 = blockIdx.y * 16;   // M tile (rows of ctx)

    const int rowA  = m0 + l;                       // A: lane = M (always valid)
    const int colB  = n0 + l;                       // B: lane = N
    const bool colOK = (colB < E);
    const int colC  = colOK ? colB : (E - 1);       // clamped, always-legal addr
    const float bm  = colOK ? 1.0f : 0.0f;          // zero-mask for OOB column

    // Per-lane base pointers; kk advances by 4 floats each iter.
    const float2* pa = (const float2*)(ctx  + (size_t)rowA * E + half * 2);
    const float2* pb = (const float2*)(Wmap + (size_t)colC * E + half * 2);

    v8f acc = {};
    for (int k = 0; k < E; k += 4) {
        const float2 af = pa[k >> 1];               // (kk, kk+1), kk = k + half*2
        const float2 bf = pb[k >> 1];
        v2f a, b;
        a.x = af.x;      a.y = af.y;
        b.x = bf.x * bm; b.y = bf.y * bm;           // cndmask/mul, no exec branch
        acc = __builtin_amdgcn_wmma_f32_16x16x4_f32(
            /*neg_a=*/false, a, /*neg_b=*/false, b,
            /*c_mod=*/(short)0, acc, /*reuse_a=*/false, /*reuse_b=*/false);
    }

    const int n = n0 + l;
    if (n < E) {
        const int mb = m0 + half * 8;
#pragma unroll
        for (int r = 0; r < 8; ++r)
            aligned[(size_t)(mb + r) * E + n] = acc[r];
    }
}

// ---------------------------------------------------------------------------
// Kernel 2: per-row top-5 + tiny attention, fused.
// One 256-thread block (8 wave32) per row b.  HBM-bound: streams 200 KB/row.
// ---------------------------------------------------------------------------
__device__ __forceinline__ void ins5(float* v, int* ix, float nv, int ni)
{
    if (!(nv > v[TK - 1])) return;           // ties keep earlier index
    int j = TK - 1;
    while (j > 0 && nv > v[j - 1]) {
        v[j] = v[j - 1]; ix[j] = ix[j - 1]; --j;
    }
    v[j] = nv; ix[j] = ni;
}

__device__ __forceinline__ void merge5(float* av, int* ai,
                                       const float* bv, const int* bi)
{
    float rv[TK]; int ri[TK];
    int p = 0, q = 0;
#pragma unroll
    for (int o = 0; o < TK; ++o) {
        bool takeA = (q >= TK) ||
                     (p < TK && (av[p] > bv[q] ||
                                 (av[p] == bv[q] && ai[p] <= bi[q])));
        if (takeA) { rv[o] = av[p]; ri[o] = ai[p]; ++p; }
        else       { rv[o] = bv[q]; ri[o] = bi[q]; ++q; }
    }
#pragma unroll
    for (int o = 0; o < TK; ++o) { av[o] = rv[o]; ai[o] = ri[o]; }
}

__global__ void __launch_bounds__(256)
topk_attn_fused(const float* __restrict__ sp_z,    // [B,Z]
                const float* __restrict__ sp_w,    // [Z,E]
                const float* __restrict__ aligned, // [B,E]
                float* __restrict__ out_v,         // [B,E]
                float* __restrict__ out_attn,      // [B,K]
                float* __restrict__ out_amax,      // [B]
                int Z, int E)
{
    const int b    = blockIdx.x;
    const int tid  = threadIdx.x;
    const int lane = tid & 31;               // wave32 lane
    const int wv   = tid >> 5;               // wave id (0..7)

    __shared__ float lv[256 * TK];
    __shared__ int   li[256 * TK];
    __shared__ int   topidx[TK];
    __shared__ float scores[TK];
    __shared__ float attn[TK];

    // ---- phase 1: streaming local top-5 (float4, HBM-bound) ----
    float v[TK]; int ix[TK];
#pragma unroll
    for (int i = 0; i < TK; ++i) { v[i] = -__builtin_inff(); ix[i] = 0x7fffffff; }

    const float4* row = (const float4*)(sp_z + (size_t)b * Z);
    const int n4 = Z >> 2;                   // 50000/4 = 12500
    for (int i = tid; i < n4; i += 256) {
        __builtin_prefetch(row + i + 512, 0, 1);   // global_prefetch_b8
        float4 q = row[i];
        const int base = i << 2;
        ins5(v, ix, q.x, base);
        ins5(v, ix, q.y, base + 1);
        ins5(v, ix, q.z, base + 2);
        ins5(v, ix, q.w, base + 3);
    }
#pragma unroll
    for (int i = 0; i < TK; ++i) { lv[tid * TK + i] = v[i]; li[tid * TK + i] = ix[i]; }
    __syncthreads();

    // ---- LDS tree merge down to one sorted top-5 ----
    for (int s = 128; s > 0; s >>= 1) {
        if (tid < s)
            merge5(&lv[tid * TK], &li[tid * TK],
                   &lv[(tid + s) * TK], &li[(tid + s) * TK]);
        __syncthreads();
    }

    if (tid < TK) { topidx[tid] = li[tid]; scores[tid] = 0.0f; }
    __syncthreads();

    // ---- phase 2: scores[k] = <sp_w[topidx[k],:], aligned[b,:]> ----
    float part[TK];
#pragma unroll
    for (int k = 0; k < TK; ++k) part[k] = 0.0f;
    for (int e = tid; e < E; e += 256) {
        const float ac = aligned[(size_t)b * E + e];
#pragma unroll
        for (int k = 0; k < TK; ++k)
            part[k] += sp_w[(size_t)topidx[k] * E + e] * ac;
    }
    // wave32 butterfly reduce, then one ds_add_f32 per wave per k
#pragma unroll
    for (int k = 0; k < TK; ++k) {
#pragma unroll
        for (int m = 16; m > 0; m >>= 1)
            part[k] += __shfl_xor(part[k], m, 32);
        if (lane == 0) atomicAdd(&scores[k], part[k]);
    }
    (void)wv;
    __syncthreads();

    // ---- phase 3: softmax + argmax (thread 0) ----
    if (tid == 0) {
        float m = scores[0];
#pragma unroll
        for (int k = 1; k < TK; ++k) m = fmaxf(m, scores[k]);
        float ssum = 0.0f; float ex[TK];
#pragma unroll
        for (int k = 0; k < TK; ++k) { ex[k] = __expf(scores[k] - m); ssum += ex[k]; }
        const float inv = 1.0f / ssum;
        int am = 0; float best = -__builtin_inff();
#pragma unroll
        for (int k = 0; k < TK; ++k) {
            attn[k] = ex[k] * inv;
            out_attn[(size_t)b * TK + k] = attn[k];
            if (attn[k] > best) { best = attn[k]; am = k; }
        }
        out_amax[b] = (float)am;
    }
    __syncthreads();

    // ---- phase 4: out[b,:] = sum_k attn[k] * sp_w[topidx[k],:] ----
    for (int e = tid; e < E; e += 256) {
        float s = 0.0f;
#pragma unroll
        for (int k = 0; k < TK; ++k)
            s += attn[k] * sp_w[(size_t)topidx[k] * E + e];
        out_v[(size_t)b * E + e] = s;
    }
}

// ---------------------------------------------------------------------------
extern "C" void kernel_launch(void* const* d_in, const int* in_sizes, int n_in,
                              void* d_out, int out_size, void* d_ws, size_t ws_size,
                              hipStream_t stream)
{
    (void)in_sizes; (void)n_in; (void)out_size; (void)d_ws; (void)ws_size;

    const float* sp_z = (const float*)d_in[0];   // [B,Z]
    const float* sp_w = (const float*)d_in[1];   // [Z,E]
    const float* ctx  = (const float*)d_in[2];   // [B,E]
    const float* Wmap = (const float*)d_in[3];   // [E,E]

    const int B = 4096, Z = 50000, E = 300, K = TK;

    float* out     = (float*)d_out;
    float* aligned = out;                                  // [B,E]
    float* out_v   = out + (size_t)B * E;                  // [B,E]
    float* out_at  = out + (size_t)2 * B * E;              // [B,K]
    float* out_am  = out_at + (size_t)B * K;               // [B]

    // Kernel 1: WMMA f32 GEMM, one wave per 16x16 tile
    dim3 g1((E + 15) / 16, B / 16);
    gemm_aligned_wmma<<<g1, 32, 0, stream>>>(ctx, Wmap, aligned, E);

    // Kernel 2: fused top-k + attention, one block per row
    topk_attn_fused<<<B, 256, 0, stream>>>(sp_z, sp_w, aligned,
                                           out_v, out_at, out_am, Z, E);
}